// MultiscaleGeodesicSolver_13314398617884
// MI455X (gfx1250) — compile-verified
//
#include <hip/hip_runtime.h>
#include <math.h>

// ---------------- problem constants ----------------
#define HFULL     512
#define NSRC      4
#define INIT_DIST 100000.0f
#define BIG       1.0e30f

// tiled solver geometry: REG x REG region = TILE interior + HALO ring.
// HALO iterations per launch are exact in the TILE interior.
#define TILE 48
#define HALO 8
#define REG  64          // TILE + 2*HALO
#define PW   66          // REG + 2 (BIG-padded frame -> branch-free 3x3 min)
#define SOLVE_BLOCK 256  // 8 wave32s

#define AS3 __attribute__((address_space(3)))
#define AS1 __attribute__((address_space(1)))

typedef __attribute__((ext_vector_type(4))) int   v4i;
typedef __attribute__((ext_vector_type(2))) float v2f;
typedef __attribute__((ext_vector_type(8))) float v8f;

// =====================================================================
// WMMA GEMM:  D = outScale * (A[M x K] * B[K x N]),  all row-major f32.
// One wave per 16x16 D tile (V_WMMA_F32_16X16X4_F32), 8 waves / block.
// Used for the separable bilinear upsample: dist_up = 2 * (U * dist * U^T).
// =====================================================================
__global__ __launch_bounds__(256)
void k_gemm_wmma(const float* __restrict__ A, const float* __restrict__ B,
                 float* __restrict__ D, int M, int N, int K,
                 int strideA, int strideB, int strideD, float outScale) {
  int wave = threadIdx.x >> 5;
  int lane = threadIdx.x & 31;
  int tilesN = N >> 4;
  int nTiles = (M >> 4) * tilesN;
  int tileId = blockIdx.x * 8 + wave;
  if (tileId >= nTiles) return;            // wave-uniform: EXEC stays all-ones
  int tm = tileId / tilesN, tn = tileId - tm * tilesN;
  int m0 = tm << 4, n0 = tn << 4;
  const float* Ab = A + (size_t)blockIdx.y * strideA;
  const float* Bb = B + (size_t)blockIdx.y * strideB;
  float*       Db = D + (size_t)blockIdx.y * strideD;
  int half = lane >> 4, ln = lane & 15;

#if __has_builtin(__builtin_amdgcn_wmma_f32_16x16x4_f32)
  v8f acc = {0.f, 0.f, 0.f, 0.f, 0.f, 0.f, 0.f, 0.f};
  for (int k = 0; k < K; k += 4) {
    int ka = k + 2 * half;                 // ISA 16x16x4 f32 A/B lane layout
    v2f a, b;
    a.x = Ab[(size_t)(m0 + ln) * K + ka];
    a.y = Ab[(size_t)(m0 + ln) * K + ka + 1];
    b.x = Bb[(size_t)ka * N + n0 + ln];
    b.y = Bb[(size_t)(ka + 1) * N + n0 + ln];
    acc = __builtin_amdgcn_wmma_f32_16x16x4_f32(false, a, false, b,
                                                (short)0, acc, false, false);
  }
#pragma unroll
  for (int g = 0; g < 8; ++g)              // C/D layout: VGPR g -> row m0+g+8*half
    Db[(size_t)(m0 + g + 8 * half) * N + n0 + ln] = outScale * acc[g];
#else
  // scalar fallback (also keeps the host pass well-formed)
  for (int g = 0; g < 8; ++g) {
    int r = m0 + g + 8 * half;
    float s = 0.f;
    for (int k = 0; k < K; ++k) s += Ab[(size_t)r * K + k] * Bb[(size_t)k * N + n0 + ln];
    Db[(size_t)r * N + n0 + ln] = outScale * s;
  }
#endif
}

// ---------------- CDNA5 async global->LDS helpers ----------------
__device__ __forceinline__ void async_ld_b32(float* lds, const float* g) {
#if __has_builtin(__builtin_amdgcn_global_load_async_to_lds_b32)
  __builtin_amdgcn_global_load_async_to_lds_b32((AS1 int*)g, (AS3 int*)lds, 0, 0);
#else
  asm volatile("global_load_async_to_lds_b32 %0, %1, off"
               :: "v"((unsigned)(unsigned long long)(void*)lds), "v"(g) : "memory");
#endif
}

__device__ __forceinline__ void async_ld_b128(float* lds, const float* g) {
#if __has_builtin(__builtin_amdgcn_global_load_async_to_lds_b128)
  __builtin_amdgcn_global_load_async_to_lds_b128((AS1 v4i*)g, (AS3 v4i*)lds, 0, 0);
#else
  asm volatile("global_load_async_to_lds_b128 %0, %1, off"
               :: "v"((unsigned)(unsigned long long)(void*)lds), "v"(g) : "memory");
#endif
}

__device__ __forceinline__ void async_wait0() {
#if __has_builtin(__builtin_amdgcn_s_wait_asynccnt)
  __builtin_amdgcn_s_wait_asynccnt(0);
#else
  asm volatile("s_wait_asynccnt 0x0" ::: "memory");
#endif
}

// one min-plus relaxation of a cell at padded index p (row pitch PW)
__device__ __forceinline__ float relax_cell(const float* cur, int p, float c) {
  float m = fminf(fminf(cur[p - PW - 1], cur[p - PW]), cur[p - PW + 1]);
  m = fminf(m, fminf(cur[p - 1], cur[p + 1]));
  float ctr = cur[p];
  m = fminf(m, ctr);
  m = fminf(m, fminf(fminf(cur[p + PW - 1], cur[p + PW]), cur[p + PW + 1]));
  return fminf(ctr, m + c);
}

// ---------------- cost pyramid ----------------
// jax.image.resize(linear, antialias=False) at scales 1/2,1/4,1/8 reduces to
// exact 2x2 averages at offsets 0/1/3 with step 2/4/8.
__global__ void k_downsample(const float* __restrict__ full, float* __restrict__ out,
                             int Ws, int off, int step) {
  int i = blockIdx.x * blockDim.x + threadIdx.x;
  if (i >= Ws * Ws) return;
  int y = i / Ws, x = i - y * Ws;
  const float* r0 = full + (y * step + off) * HFULL + (x * step + off);
  const float* r1 = r0 + HFULL;
  out[i] = 0.25f * (r0[0] + r0[1] + r1[0] + r1[1]);
}

// ---------------- build 2x bilinear interpolation matrices U / U^T -------
// row c of U: weights (1-w), w at clamped cols j0, j0+1; fy = 0.5c - 0.25.
__global__ void k_zero(float* __restrict__ p, int n) {
  int i = blockIdx.x * blockDim.x + threadIdx.x;
  if (i < n) p[i] = 0.f;
}

__global__ void k_build_U(float* __restrict__ U, float* __restrict__ UT,
                          int Wout, int Win) {
  int c = blockIdx.x * blockDim.x + threadIdx.x;
  if (c >= Wout) return;
  float f = (c + 0.5f) * 0.5f - 0.5f;
  int j0 = (int)floorf(f);
  float w = f - (float)j0;
  int j0c = j0 < 0 ? 0 : j0;
  int j1c = (j0 + 1 >= Win) ? Win - 1 : j0 + 1;
  U[(size_t)c * Win + j0c] += 1.0f - w;
  U[(size_t)c * Win + j1c] += w;
  UT[(size_t)j0c * Wout + c] += 1.0f - w;   // distinct (row,col) per thread: race-free
  UT[(size_t)j1c * Wout + c] += w;
}

// ---------------- pin sources to 0 ----------------
__global__ void k_set_sources(float* __restrict__ dist, const int* __restrict__ sc,
                              int Ws, float scale) {
  int n = threadIdx.x;
  if (n >= NSRC) return;
  int sx = (int)((float)sc[2 * n]     * scale);
  int sy = (int)((float)sc[2 * n + 1] * scale);
  sx = sx < 0 ? 0 : (sx > Ws - 1 ? Ws - 1 : sx);
  sy = sy < 0 ? 0 : (sy > Ws - 1 ? Ws - 1 : sy);
  dist[(size_t)n * Ws * Ws + (size_t)sy * Ws + sx] = 0.0f;
}

// ---------------- coarsest scale: whole 64x64 map LDS-resident, 1000 iters ----
__global__ __launch_bounds__(1024)
void k_solve64(const float* __restrict__ cost64, const int* __restrict__ sc,
               float* __restrict__ dout, int iters) {
  __shared__ __align__(16) float cs[REG * REG];
  __shared__ __align__(16) float bufA[PW * PW];
  __shared__ __align__(16) float bufB[PW * PW];
  int tid = threadIdx.x;
  int n = blockIdx.x;

  for (int i = tid; i < PW; i += blockDim.x) {     // BIG frame (min-pool +inf pad)
    bufA[i] = BIG; bufA[(PW - 1) * PW + i] = BIG;
    bufB[i] = BIG; bufB[(PW - 1) * PW + i] = BIG;
    bufA[i * PW] = BIG; bufA[i * PW + PW - 1] = BIG;
    bufB[i * PW] = BIG; bufB[i * PW + PW - 1] = BIG;
  }
  for (int c = tid; c < REG * REG; c += blockDim.x) {
    int ry = c >> 6, rx = c & 63;
    bufA[(ry + 1) * PW + rx + 1] = INIT_DIST;
  }
  { // async DMA the 16KB cost tile into LDS, 16B chunks
    int c4 = tid * 4;
    if (c4 < REG * REG) async_ld_b128(&cs[c4], &cost64[c4]);
  }
  async_wait0();
  __syncthreads();
  if (tid == 0) {
    int sx = (int)((float)sc[2 * n] * 0.125f);
    int sy = (int)((float)sc[2 * n + 1] * 0.125f);
    sx = sx < 0 ? 0 : (sx > REG - 1 ? REG - 1 : sx);
    sy = sy < 0 ? 0 : (sy > REG - 1 ? REG - 1 : sy);
    bufA[(sy + 1) * PW + sx + 1] = 0.0f;
  }
  __syncthreads();

  float* cur = bufA; float* nxt = bufB;
  for (int it = 0; it < iters; ++it) {
    for (int c = tid; c < REG * REG; c += blockDim.x) {
      int ry = c >> 6, rx = c & 63;
      int p = (ry + 1) * PW + rx + 1;
      nxt[p] = relax_cell(cur, p, cs[c]);
    }
    __syncthreads();
    float* t = cur; cur = nxt; nxt = t;
  }
  float* g = dout + (size_t)n * REG * REG;
  for (int c = tid; c < REG * REG; c += blockDim.x) {
    int ry = c >> 6, rx = c & 63;
    g[c] = cur[(ry + 1) * PW + rx + 1];
  }
}

// ---------------- tiled halo solver: k (<=HALO) exact Jacobi iters in LDS ----
__global__ __launch_bounds__(SOLVE_BLOCK)
void k_solve_tile(const float* __restrict__ din, const float* __restrict__ cost,
                  float* __restrict__ dout, int Ws, int nTx, int kIters) {
  __shared__ __align__(16) float cs[REG * REG];
  __shared__ __align__(16) float bufA[PW * PW];
  __shared__ __align__(16) float bufB[PW * PW];
  int tid = threadIdx.x;
  int tIdx = blockIdx.x;
  int n = blockIdx.y;
  int tx0 = (tIdx % nTx) * TILE - HALO;
  int ty0 = (tIdx / nTx) * TILE - HALO;
  const float* dbase = din + (size_t)n * Ws * Ws;
  float* obase = dout + (size_t)n * Ws * Ws;

  // hint the write-back region into cache early
  for (int iy = tid; iy < TILE; iy += SOLVE_BLOCK) {
    int gy = ty0 + HALO + iy;
    if (gy < Ws) __builtin_prefetch(&obase[(size_t)gy * Ws + (tx0 + HALO)], 1, 1);
  }

  for (int i = tid; i < PW; i += SOLVE_BLOCK) {
    bufA[i] = BIG; bufA[(PW - 1) * PW + i] = BIG;
    bufB[i] = BIG; bufB[(PW - 1) * PW + i] = BIG;
    bufA[i * PW] = BIG; bufA[i * PW + PW - 1] = BIG;
    bufB[i * PW] = BIG; bufB[i * PW + PW - 1] = BIG;
  }

  // region fill: 64 rows x 16 four-float chunks; async DMA where fully in-grid
  for (int ch = tid; ch < REG * (REG / 4); ch += SOLVE_BLOCK) {
    int ry  = ch / (REG / 4);
    int rx0 = (ch % (REG / 4)) * 4;
    int gy  = ty0 + ry;
    int gx0 = tx0 + rx0;
    int p   = (ry + 1) * PW + rx0 + 1;
    bool rowOk = (gy >= 0) && (gy < Ws);
    if (rowOk && gx0 >= 0 && gx0 + 3 < Ws) {
      const float* g = dbase + (size_t)gy * Ws + gx0;
      async_ld_b32(&bufA[p],     g);
      async_ld_b32(&bufA[p + 1], g + 1);
      async_ld_b32(&bufA[p + 2], g + 2);
      async_ld_b32(&bufA[p + 3], g + 3);
      async_ld_b128(&cs[ry * REG + rx0], cost + (size_t)gy * Ws + gx0);  // gx0 % 4 == 0
    } else {
      for (int j = 0; j < 4; ++j) {
        int gx = gx0 + j;
        bool ok = rowOk && gx >= 0 && gx < Ws;
        bufA[p + j]            = ok ? dbase[(size_t)gy * Ws + gx] : BIG;  // dead cells stay BIG
        cs[ry * REG + rx0 + j] = ok ? cost[(size_t)gy * Ws + gx]  : BIG;
      }
    }
  }
  async_wait0();
  __syncthreads();

  float* cur = bufA; float* nxt = bufB;
  for (int it = 0; it < kIters; ++it) {
    for (int c = tid; c < REG * REG; c += SOLVE_BLOCK) {
      int ry = c >> 6, rx = c & 63;
      int p = (ry + 1) * PW + rx + 1;
      nxt[p] = relax_cell(cur, p, cs[c]);
    }
    __syncthreads();
    float* t = cur; cur = nxt; nxt = t;
  }

  // interior write-back (exact for kIters <= HALO)
  for (int c = tid; c < TILE * TILE; c += SOLVE_BLOCK) {
    int iy = c / TILE, ix = c - iy * TILE;
    int gy = ty0 + HALO + iy, gx = tx0 + HALO + ix;
    if (gy < Ws && gx < Ws)
      obase[(size_t)gy * Ws + gx] = cur[(HALO + iy + 1) * PW + (HALO + ix + 1)];
  }
}

// ---------------- host orchestration ----------------
extern "C" void kernel_launch(void* const* d_in, const int* in_sizes, int n_in,
                              void* d_out, int out_size, void* d_ws, size_t ws_size,
                              hipStream_t stream) {
  const float* cost_full = (const float*)d_in[0];
  const int*   sc        = (const int*)d_in[1];

  float* ws   = (float*)d_ws;
  float* c64  = ws;                                   // 64*64
  float* c128 = c64  + 64 * 64;                       // 128*128
  float* c256 = c128 + 128 * 128;                     // 256*256
  float* bufA = c256 + 256 * 256;                     // 4*512*512
  float* bufB = bufA + (size_t)NSRC * 512 * 512;      // 4*512*512
  float* U128  = bufB + (size_t)NSRC * 512 * 512;     // 128*64
  float* UT128 = U128  + 128 * 64;                    // 64*128
  float* U256  = UT128 + 128 * 64;                    // 256*128
  float* UT256 = U256  + 256 * 128;                   // 128*256
  float* U512  = UT256 + 256 * 128;                   // 512*256
  float* UT512 = U512  + 512 * 256;                   // 256*512
  float* T1    = UT512 + 512 * 256;                   // 4 * 256*512 (x-interp temp)
  const int uTotal = 2 * (128 * 64 + 256 * 128 + 512 * 256);

  // cost pyramid (exact closed-form bilinear at these scales)
  k_downsample<<<(64 * 64 + 255) / 256,   256, 0, stream>>>(cost_full, c64,  64,  3, 8);
  k_downsample<<<(128 * 128 + 255) / 256, 256, 0, stream>>>(cost_full, c128, 128, 1, 4);
  k_downsample<<<(256 * 256 + 255) / 256, 256, 0, stream>>>(cost_full, c256, 256, 0, 2);

  // interpolation matrices for the WMMA upsample (rebuilt every call: no cached state)
  k_zero<<<(uTotal + 255) / 256, 256, 0, stream>>>(U128, uTotal);
  k_build_U<<<1, 128, 0, stream>>>(U128, UT128, 128, 64);
  k_build_U<<<1, 256, 0, stream>>>(U256, UT256, 256, 128);
  k_build_U<<<2, 256, 0, stream>>>(U512, UT512, 512, 256);

  // scale 0.125: 1000 iterations fully LDS-resident (one WG of 32 waves / map)
  k_solve64<<<NSRC, 1024, 0, stream>>>(c64, sc, bufA, 1000);

  float* cur = bufA;
  const float* costs[3] = {c128, c256, cost_full};
  float* Us[3]  = {U128, U256, U512};
  float* UTs[3] = {UT128, UT256, UT512};
  const int   sizes[3] = {128, 256, 512};
  const float scl[3]   = {0.25f, 0.5f, 1.0f};

  for (int s = 0; s < 3; ++s) {
    int Ws = sizes[s];
    int Win = Ws / 2;
    float* oth = (cur == bufA) ? bufB : bufA;

    // dist_up = 2 * (U * dist * U^T) via two WMMA GEMMs (x-interp, then y-interp)
    {
      int t1Tiles = (Win >> 4) * (Ws >> 4);           // T1 = dist(Win x Win) * UT(Win x Ws)
      dim3 g1((t1Tiles + 7) / 8, NSRC);
      k_gemm_wmma<<<g1, 256, 0, stream>>>(cur, UTs[s], T1, Win, Ws, Win,
                                          Win * Win, 0, Win * Ws, 1.0f);
      int t2Tiles = (Ws >> 4) * (Ws >> 4);            // out = 2 * U(Ws x Win) * T1(Win x Ws)
      dim3 g2((t2Tiles + 7) / 8, NSRC);
      k_gemm_wmma<<<g2, 256, 0, stream>>>(Us[s], T1, oth, Ws, Ws, Win,
                                          0, Win * Ws, Ws * Ws, 2.0f);
    }
    cur = oth;
    k_set_sources<<<1, NSRC, 0, stream>>>(cur, sc, Ws, scl[s]);

    int nTx = (Ws + TILE - 1) / TILE;
    dim3 grid(nTx * nTx, NSRC);
    int rem = 100;
    while (rem > 0) {
      int k = rem > HALO ? HALO : rem;
      rem -= k;
      float* nb  = (cur == bufA) ? bufB : bufA;
      float* out = (s == 2 && rem == 0) ? (float*)d_out : nb;  // final pass lands in d_out
      k_solve_tile<<<grid, SOLVE_BLOCK, 0, stream>>>(cur, costs[s], out, Ws, nTx, k);
      cur = out;
    }
  }
  (void)in_sizes; (void)n_in; (void)out_size; (void)ws_size;
}